// GNNRegressor_67765993997195
// MI455X (gfx1250) — compile-verified
//
#include <hip/hip_runtime.h>
#include <cfloat>
#include <math.h>

// ---------------------------------------------------------------------------
// GNNRegressor forward for MI455X (gfx1250, wave32).
//  - Dense GEMMs via V_WMMA_F32_16X16X4_F32 (fp32-exact, matmul is <<1% of cost)
//  - GCN propagation: precomputed sym-norm, wave-per-edge coalesced atomics
// ---------------------------------------------------------------------------

typedef float v2f __attribute__((ext_vector_type(2)));
typedef float v8f __attribute__((ext_vector_type(8)));

// ---------------- utility kernels ----------------

__global__ __launch_bounds__(256) void zero_kernel(float* __restrict__ p, int n) {
  int i = blockIdx.x * blockDim.x + threadIdx.x;
  if (i < n) p[i] = 0.0f;
}

__global__ __launch_bounds__(256) void fill_kernel(float* __restrict__ p, float v, int n) {
  int i = blockIdx.x * blockDim.x + threadIdx.x;
  if (i < n) p[i] = v;
}

// in-degree count over dst (self-loop added later as +1)
__global__ __launch_bounds__(256) void deg_kernel(const int* __restrict__ dst,
                                                  float* __restrict__ deg, int E) {
  int e = blockIdx.x * blockDim.x + threadIdx.x;
  if (e < E) atomicAdd(&deg[dst[e]], 1.0f);
}

// deg -> deg^{-1/2}, in place (self loop => deg+1 >= 1, never zero)
__global__ __launch_bounds__(256) void dinv_kernel(float* __restrict__ deg, int n) {
  int i = blockIdx.x * blockDim.x + threadIdx.x;
  if (i < n) deg[i] = rsqrtf(deg[i] + 1.0f);
}

// per-edge symmetric normalization: norm[e] = dinv[src]*dinv[dst]
__global__ __launch_bounds__(256) void norm_kernel(const int* __restrict__ src,
                                                   const int* __restrict__ dst,
                                                   const float* __restrict__ dinv,
                                                   float* __restrict__ norm, int E) {
  int e = blockIdx.x * blockDim.x + threadIdx.x;
  if (e < E) norm[e] = dinv[src[e]] * dinv[dst[e]];
}

// ---------------- WMMA GEMM:  C[N,F] = H[N,K] @ W[K,F]  (fp32) ----------------
// One wave32 per 16x16 output tile, K-loop in steps of 4 using
// V_WMMA_F32_16X16X4_F32.  Fragment layouts per CDNA5 ISA 7.12.2:
//   A (16x4 fp32):  lanes 0-15 -> row M=lane, v0=K0 v1=K1; lanes 16-31 -> K2,K3
//   B (4x16 fp32):  lanes 0-15 -> col N=lane, v0=K0 v1=K1; lanes 16-31 -> K2,K3
//   C/D (16x16):    vgpr r: lanes 0-15 -> M=r, lanes 16-31 -> M=r+8; N=lane&15
// nrows must be a multiple of 16 (N=100000 = 6250*16); tile predicates are
// wave-uniform so EXEC is all-ones inside every wave that reaches the WMMA.
template <int K, int F>
__global__ __launch_bounds__(256) void gemm_wmma_kernel(const float* __restrict__ H,
                                                        const float* __restrict__ W,
                                                        float* __restrict__ C,
                                                        int nrows) {
  const int lane   = threadIdx.x & 31;
  const int wave   = blockIdx.x * (blockDim.x >> 5) + (threadIdx.x >> 5);
  const int tilesN = F / 16;
  const int tileM  = (wave / tilesN) * 16;
  const int tileN  = (wave % tilesN) * 16;
  if (tileM >= nrows) return;  // wave-uniform exit

  const int l15 = lane & 15;
  const int hi  = lane >> 4;  // 0: K=0,1   1: K=2,3

  const float* __restrict__ arow = H + (size_t)(tileM + l15) * K + (hi * 2);
  const float* __restrict__ bcol = W + (size_t)(hi * 2) * F + tileN + l15;

  v8f acc = {};
  for (int k = 0; k < K; k += 4) {
    v2f a, b;
    a.x = arow[k];
    a.y = arow[k + 1];
    b.x = bcol[(size_t)k * F];
    b.y = bcol[(size_t)(k + 1) * F];
    // 8-arg form: (neg_a, A, neg_b, B, c_mod, C, reuse_a, reuse_b)
    acc = __builtin_amdgcn_wmma_f32_16x16x4_f32(false, a, false, b, (short)0,
                                                acc, false, false);
  }

  float* __restrict__ crow = C + (size_t)(tileM + hi * 8) * F + tileN + l15;
#pragma unroll
  for (int r = 0; r < 8; ++r) crow[(size_t)r * F] = acc[r];
}

// ---------------- propagation ----------------

// self-loop term, also initializes the output buffer:
//   out[i,f] = in[i,f] * dinv[i]^2
template <int F>
__global__ __launch_bounds__(256) void selfloop_kernel(const float* __restrict__ in,
                                                       float* __restrict__ out,
                                                       const float* __restrict__ dinv,
                                                       int total) {
  int idx = blockIdx.x * blockDim.x + threadIdx.x;
  if (idx < total) {
    int row = idx / F;  // F is a power of two -> shift
    float d = dinv[row];
    out[idx] = in[idx] * d * d;
  }
}

// edge scatter: one wave per edge; lanes stride the F contiguous features.
// src/dst/norm are wave-uniform -> scalar loads; feature traffic is coalesced.
template <int F>
__global__ __launch_bounds__(256) void prop_edges_kernel(const float* __restrict__ in,
                                                         float* __restrict__ out,
                                                         const int* __restrict__ src,
                                                         const int* __restrict__ dst,
                                                         const float* __restrict__ norm,
                                                         int E) {
  int e = blockIdx.x * (blockDim.x >> 5) + (threadIdx.x >> 5);
  if (e >= E) return;
  int lane = threadIdx.x & 31;
  int s = src[e];
  int d = dst[e];
  float w = norm[e];
  const float* __restrict__ pin = in + (size_t)s * F;
  float* __restrict__ pout = out + (size_t)d * F;
  __builtin_prefetch(pin, 0, 0);
#pragma unroll
  for (int f = lane; f < F; f += 32) atomicAdd(&pout[f], pin[f] * w);
}

// bias (+ optional relu) in place
template <int F, bool RELU>
__global__ __launch_bounds__(256) void bias_act_kernel(float* __restrict__ buf,
                                                       const float* __restrict__ b,
                                                       int total) {
  int idx = blockIdx.x * blockDim.x + threadIdx.x;
  if (idx < total) {
    float v = buf[idx] + b[idx % F];
    buf[idx] = RELU ? fmaxf(v, 0.0f) : v;
  }
}

// ---------------- pooling + MLP ----------------

__device__ __forceinline__ void atomicMaxFloat(float* addr, float val) {
  // monotone int/uint encoding trick (correct for +0, inf, finite values; -0 hits the signed branch as INT_MIN and never replaces a negative)
  if (val >= 0.0f)
    atomicMax((int*)addr, __float_as_int(val));
  else
    atomicMin((unsigned int*)addr, (unsigned int)__float_as_int(val));
}

__global__ __launch_bounds__(256) void segmax_kernel(const float* __restrict__ z,
                                                     const int* __restrict__ batch,
                                                     float* __restrict__ pool,
                                                     int nnodes) {
  int idx = blockIdx.x * blockDim.x + threadIdx.x;
  if (idx < nnodes * 64) {
    int i = idx >> 6;
    int f = idx & 63;
    int g = batch[i];
    atomicMaxFloat(&pool[g * 64 + f], z[idx]);
  }
}

__global__ __launch_bounds__(256) void mlp_kernel(const float* __restrict__ pool,
                                                  const float* __restrict__ fc1_w,
                                                  const float* __restrict__ fc1_b,
                                                  const float* __restrict__ fc2_w,
                                                  const float* __restrict__ fc2_b,
                                                  const float* __restrict__ cpd_w,
                                                  const float* __restrict__ cpd_b,
                                                  const float* __restrict__ comb_w,
                                                  const float* __restrict__ comb_b,
                                                  float* __restrict__ out) {
  int g = threadIdx.x;  // one graph per lane, single block of 256
  if (g >= 256) return;
  float zin[64];
#pragma unroll
  for (int i = 0; i < 64; ++i) zin[i] = pool[g * 64 + i];
  float h1[32];
#pragma unroll 4
  for (int o = 0; o < 32; ++o) {
    float s = fc1_b[o];
    for (int i = 0; i < 64; ++i) s += zin[i] * fc1_w[i * 32 + o];
    h1[o] = fmaxf(s, 0.0f);
  }
  float h2[16];
#pragma unroll 4
  for (int o = 0; o < 16; ++o) {
    float s = fc2_b[o];
    for (int i = 0; i < 32; ++i) s += h1[i] * fc2_w[i * 16 + o];
    h2[o] = fmaxf(s, 0.0f);
  }
  float cpd = cpd_b[0], comb = comb_b[0];
#pragma unroll
  for (int i = 0; i < 16; ++i) {
    cpd += h2[i] * cpd_w[i];
    comb += h2[i] * comb_w[i];
  }
  out[g] = cpd;        // cpd_pred, flattened first
  out[256 + g] = comb; // comb_logits
}

// ---------------- host orchestration ----------------

static inline int cdiv_i(long long a, long long b) { return (int)((a + b - 1) / b); }

extern "C" void kernel_launch(void* const* d_in, const int* in_sizes, int n_in,
                              void* d_out, int out_size, void* d_ws, size_t ws_size,
                              hipStream_t stream) {
  const float* x       = (const float*)d_in[0];
  const int*   eidx    = (const int*)d_in[1];   // [2,E] int32 (jax default x64 off)
  const int*   batch   = (const int*)d_in[2];   // [N]
  const float* enc_w1  = (const float*)d_in[3];
  const float* enc_b1  = (const float*)d_in[4];
  const float* enc_w2  = (const float*)d_in[5];
  const float* enc_b2  = (const float*)d_in[6];
  const float* w1      = (const float*)d_in[7];
  const float* b1      = (const float*)d_in[8];
  const float* w2      = (const float*)d_in[9];
  const float* b2      = (const float*)d_in[10];
  const float* w3      = (const float*)d_in[11];
  const float* b3      = (const float*)d_in[12];
  const float* sg_w    = (const float*)d_in[13];
  const float* sg_b    = (const float*)d_in[14];
  const float* fc1_w   = (const float*)d_in[15];
  const float* fc1_b   = (const float*)d_in[16];
  const float* fc2_w   = (const float*)d_in[17];
  const float* fc2_b   = (const float*)d_in[18];
  const float* cpd_w   = (const float*)d_in[19];
  const float* cpd_b   = (const float*)d_in[20];
  const float* comb_w  = (const float*)d_in[21];
  const float* comb_b  = (const float*)d_in[22];

  const int Nn = in_sizes[0] / 128;   // 100000
  const int Ee = in_sizes[1] / 2;     // 3200000
  const int* src = eidx;              // edge_index[0]
  const int* dst = eidx + Ee;         // edge_index[1]

  // workspace layout (floats, 256B-aligned chunks)
  size_t off = 0;
  float* base = (float*)d_ws;
  auto walloc = [&](size_t nf) {
    float* p = base + off;
    off += (nf + 63) & ~(size_t)63;
    return p;
  };
  float* dinv = walloc(Nn);                 // degree -> deg^{-1/2}
  float* norm = walloc(Ee);                 // per-edge sym norm
  float* bufA = walloc((size_t)Nn * 128);   // ping
  float* bufB = walloc((size_t)Nn * 128);   // pong
  float* pool = walloc(256 * 64);           // graph pooling

  const int TB = 256;

  // --- normalization constants (recomputed each call; ws is poisoned) ---
  zero_kernel<<<cdiv_i(Nn, TB), TB, 0, stream>>>(dinv, Nn);
  deg_kernel<<<cdiv_i(Ee, TB), TB, 0, stream>>>(dst, dinv, Ee);
  dinv_kernel<<<cdiv_i(Nn, TB), TB, 0, stream>>>(dinv, Nn);
  norm_kernel<<<cdiv_i(Ee, TB), TB, 0, stream>>>(src, dst, dinv, norm, Ee);

  auto gemm128x128 = [&](const float* H, const float* W, float* C) {
    int waves = (Nn / 16) * (128 / 16);
    gemm_wmma_kernel<128, 128><<<cdiv_i((long long)waves * 32, TB), TB, 0, stream>>>(H, W, C, Nn);
  };
  auto gemm128x64 = [&](const float* H, const float* W, float* C) {
    int waves = (Nn / 16) * (64 / 16);
    gemm_wmma_kernel<128, 64><<<cdiv_i((long long)waves * 32, TB), TB, 0, stream>>>(H, W, C, Nn);
  };
  auto gemm64x64 = [&](const float* H, const float* W, float* C) {
    int waves = (Nn / 16) * (64 / 16);
    gemm_wmma_kernel<64, 64><<<cdiv_i((long long)waves * 32, TB), TB, 0, stream>>>(H, W, C, Nn);
  };
  auto prop128 = [&](const float* in, float* out) {
    selfloop_kernel<128><<<cdiv_i((long long)Nn * 128, TB), TB, 0, stream>>>(in, out, dinv, Nn * 128);
    prop_edges_kernel<128><<<cdiv_i(Ee, TB / 32), TB, 0, stream>>>(in, out, src, dst, norm, Ee);
  };
  auto prop64 = [&](const float* in, float* out) {
    selfloop_kernel<64><<<cdiv_i((long long)Nn * 64, TB), TB, 0, stream>>>(in, out, dinv, Nn * 64);
    prop_edges_kernel<64><<<cdiv_i(Ee, TB / 32), TB, 0, stream>>>(in, out, src, dst, norm, Ee);
  };

  // --- encoder ---
  gemm128x128(x, enc_w1, bufA);                       // x @ enc_w1
  prop128(bufA, bufB);                                // aggregate
  bias_act_kernel<128, true><<<cdiv_i((long long)Nn * 128, TB), TB, 0, stream>>>(bufB, enc_b1, Nn * 128);

  gemm128x64(bufB, enc_w2, bufA);                     // z @ enc_w2
  prop64(bufA, bufB);
  bias_act_kernel<64, false><<<cdiv_i((long long)Nn * 64, TB), TB, 0, stream>>>(bufB, enc_b2, Nn * 64);

  // --- 3x GCNConv(64,64) + relu ---
  const float* Ws[3] = {w1, w2, w3};
  const float* Bs[3] = {b1, b2, b3};
  for (int l = 0; l < 3; ++l) {
    gemm64x64(bufB, Ws[l], bufA);
    prop64(bufA, bufB);
    bias_act_kernel<64, true><<<cdiv_i((long long)Nn * 64, TB), TB, 0, stream>>>(bufB, Bs[l], Nn * 64);
  }

  // --- SGConv K=4: 4 propagations, then linear ---
  prop64(bufB, bufA);
  prop64(bufA, bufB);
  prop64(bufB, bufA);
  prop64(bufA, bufB);
  gemm64x64(bufB, sg_w, bufA);
  bias_act_kernel<64, false><<<cdiv_i((long long)Nn * 64, TB), TB, 0, stream>>>(bufA, sg_b, Nn * 64);

  // --- global max pool per graph ---
  fill_kernel<<<cdiv_i(256 * 64, TB), TB, 0, stream>>>(pool, -INFINITY, 256 * 64);
  segmax_kernel<<<cdiv_i((long long)Nn * 64, TB), TB, 0, stream>>>(bufA, batch, pool, Nn);

  // --- head MLP + outputs ---
  mlp_kernel<<<1, 256, 0, stream>>>(pool, fc1_w, fc1_b, fc2_w, fc2_b,
                                    cpd_w, cpd_b, comb_w, comb_b, (float*)d_out);
}